// CrossModalHypergraphPerception_16080357556291
// MI455X (gfx1250) — compile-verified
//
#include <hip/hip_runtime.h>

// ---------------------------------------------------------------------------
// CrossModalHypergraphPerception on gfx1250 (MI455X), fp32 WMMA pipeline.
// B=4, C=256, H=W=64 -> N=4096 targets, Nc=8192 context nodes, k=8.
// Dominant work: Xt @ Xc^T (68.7 GFLOP fp32) -> V_WMMA_F32_16X16X4_F32.
// kNN sweep split 8-way along the context dimension for SIMD occupancy
// (4096 single-wave blocks instead of 512), then an 8-way top-8 merge.
// ---------------------------------------------------------------------------

// Device-pass-only feature check (host pass doesn't know amdgcn builtins).
#if defined(__gfx1250__) && !__has_builtin(__builtin_amdgcn_wmma_f32_16x16x4_f32)
#error "missing __builtin_amdgcn_wmma_f32_16x16x4_f32 on this toolchain"
#endif

typedef float v2f __attribute__((ext_vector_type(2)));
typedef float v8f __attribute__((ext_vector_type(8)));

#define BATCH  4
#define CCH    256
#define NT     4096   // target nodes per batch (64*64)
#define NC     8192   // context nodes per batch (2*N)
#define KNN    8
#define CHUNKS 8      // column-sweep split factor for the kNN kernel

static __device__ __forceinline__ v8f wmma4(v2f a, v2f b, v8f c) {
#if defined(__gfx1250__)
  // D = A(16x4 f32) * B(4x16 f32) + C(16x16 f32); 8-arg VOP3P form.
  return __builtin_amdgcn_wmma_f32_16x16x4_f32(false, a, false, b, (short)0, c,
                                               false, false);
#else
  (void)a; (void)b;
  return c;  // host pass / non-gfx1250: never executed
#endif
}

#define TSTRIDE 260  // 256 + 4 floats pad: kills LDS bank conflicts, keeps 16B align

// Stage nrows x 256 fp32 rows (row-major, stride 256) into LDS (stride 260).
static __device__ __forceinline__ void stage_tile(float* __restrict__ dst,
                                                  const float* __restrict__ src,
                                                  int nrows, int lane) {
  const int total = nrows * 256;
  for (int e = lane * 4; e < total; e += 128) {
    const int row = e >> 8, col = e & 255;
    float4 v = *(const float4*)(src + row * 256 + col);
    *(float4*)(dst + row * TSTRIDE + col) = v;
  }
}

// Ascending-sorted top-8 insertion; strict '<' keeps earliest-inserted entry
// on value ties (=> lowest index wins when candidates arrive index-ascending).
static __device__ __forceinline__ void topk_insert(float v, int idx,
                                                   float (&best)[KNN],
                                                   int (&bidx)[KNN]) {
  if (v < best[KNN - 1]) {
    best[KNN - 1] = v;
    bidx[KNN - 1] = idx;
#pragma unroll
    for (int t = KNN - 1; t > 0; --t) {
      if (best[t] < best[t - 1]) {
        float tv = best[t]; best[t] = best[t - 1]; best[t - 1] = tv;
        int ti = bidx[t];  bidx[t] = bidx[t - 1];  bidx[t - 1] = ti;
      }
    }
  }
}

// ---------------------------------------------------------------------------
// Tiled transposes (image [B,C,Nn] <-> nodes [B,Nn,C]) through LDS.
// ---------------------------------------------------------------------------
__global__ __launch_bounds__(256) void img2node_kernel(
    const float* __restrict__ in, float* __restrict__ out, int C, int Nn,
    long ibs, long obs) {
  __shared__ float tile[32][33];
  const int b = blockIdx.z;
  const int n0 = blockIdx.x * 32, c0 = blockIdx.y * 32;
  const int tx = threadIdx.x, ty = threadIdx.y;
  const float* ib = in + (long)b * ibs;
  float* ob = out + (long)b * obs;
#pragma unroll
  for (int r = 0; r < 32; r += 8)
    tile[ty + r][tx] = ib[(long)(c0 + ty + r) * Nn + n0 + tx];
  __syncthreads();
#pragma unroll
  for (int r = 0; r < 32; r += 8)
    ob[(long)(n0 + ty + r) * C + c0 + tx] = tile[tx][ty + r];
}

// nodes [rows,256] -> image [B,256,4096]; optional per-node 1/max(count,1).
__global__ __launch_bounds__(256) void node2img_kernel(
    const float* __restrict__ in, float* __restrict__ out,
    const float* __restrict__ counts, long ibs, long obs, long cbs) {
  __shared__ float tile[32][33];
  const int b = blockIdx.z;
  const int n0 = blockIdx.x * 32, c0 = blockIdx.y * 32;
  const int tx = threadIdx.x, ty = threadIdx.y;
  const float* ib = in + (long)b * ibs;
  float* ob = out + (long)b * obs;
#pragma unroll
  for (int r = 0; r < 32; r += 8) {
    const int n = n0 + ty + r;
    float sc = 1.0f;
    if (counts) sc = 1.0f / fmaxf(counts[(long)b * cbs + n], 1.0f);
    tile[ty + r][tx] = ib[(long)n * CCH + c0 + tx] * sc;
  }
  __syncthreads();
#pragma unroll
  for (int r = 0; r < 32; r += 8)
    ob[(long)(c0 + ty + r) * NT + n0 + tx] = tile[tx][ty + r];
}

// ---------------------------------------------------------------------------
// Squared norms of context rows (for d2 ordering; target norms cancel).
// ---------------------------------------------------------------------------
__global__ __launch_bounds__(256) void norms_kernel(const float* __restrict__ x,
                                                    float* __restrict__ out,
                                                    int rows) {
  const int wid = (int)((blockIdx.x * blockDim.x + threadIdx.x) >> 5);
  const int lane = threadIdx.x & 31;
  if (wid >= rows) return;
  const float* r = x + (long)wid * 256;
  float s = 0.0f;
#pragma unroll
  for (int e = 0; e < 256; e += 32) {
    float v = r[e + lane];
    s += v * v;
  }
#pragma unroll
  for (int off = 16; off > 0; off >>= 1) s += __shfl_xor(s, off, 32);
  if (lane == 0) out[wid] = s;
}

// ---------------------------------------------------------------------------
// GEMM + bias: Y[rows,256] = X[rows,256] @ W + b, W given transposed
// (WT[c_out][c_in]).  One wave per block computes a 32x16 output tile with
// 128 v_wmma_f32_16x16x4_f32.  grid = (rows/32, 16).
// ---------------------------------------------------------------------------
__global__ __launch_bounds__(32) void gemm_bias_kernel(
    const float* __restrict__ X, const float* __restrict__ WT,
    const float* __restrict__ bias, float* __restrict__ Y) {
  __shared__ float smem[32 * TSTRIDE + 16 * TSTRIDE];  // 48.8 KB
  float* al = smem;                 // A tile: 32 rows x 256 (stride 260)
  float* wl = smem + 32 * TSTRIDE;  // B tile: 16 WT rows x 256

  const int lane = threadIdx.x & 31;
  const int row0 = blockIdx.x * 32;
  const int col0 = blockIdx.y * 16;

  stage_tile(al, X + (long)row0 * 256, 32, lane);
  stage_tile(wl, WT + (long)col0 * 256, 16, lane);

  const int hw = lane >> 4, lo = lane & 15;
  const int off0 = lo * TSTRIDE + hw * 2;
  const int off1 = (lo + 16) * TSTRIDE + hw * 2;

  v8f c0 = {}, c1 = {};
#pragma unroll 8
  for (int k = 0; k < 256; k += 4) {
    v2f bv = *(const v2f*)(wl + off0 + k);   // B[k..][n=lo] = WT[col0+lo][k..]
    v2f a0 = *(const v2f*)(al + off0 + k);   // rows 0..15
    v2f a1 = *(const v2f*)(al + off1 + k);   // rows 16..31
    c0 = wmma4(a0, bv, c0);
    c1 = wmma4(a1, bv, c1);
  }

  const float bz = bias[col0 + lo];
  float* y = Y + (long)row0 * 256;
#pragma unroll
  for (int r = 0; r < 8; ++r) {
    const int m0 = r + hw * 8;  // D layout: vgpr r, half-wave -> row
    y[(long)m0 * 256 + col0 + lo] = c0[r] + bz;
    y[(long)(m0 + 16) * 256 + col0 + lo] = c1[r] + bz;
  }
}

// ---------------------------------------------------------------------------
// kNN partial sweep: per wave, 32 target rows vs a 1/CHUNKS slice of the
// 8192 context nodes.  score(i,j) = ||c_j||^2 - 2 <t_i, c_j>  (same argmins
// as full d2; target norms cancel).  64 column tiles x 128 WMMA per wave;
// per-lane register top-8 -> global partial lists.
// grid = (B*NT/32, CHUNKS) single-wave blocks (4096 waves).
// ---------------------------------------------------------------------------
__global__ __launch_bounds__(32) void knn_sweep_kernel(
    const float* __restrict__ xt, const float* __restrict__ xc,
    const float* __restrict__ nc, float* __restrict__ pval,
    int* __restrict__ pidx) {
  __shared__ float smem[32 * TSTRIDE + 16 * TSTRIDE + 32 * 20];  // 51.3 KB
  float* at = smem;                                // 32 x 260 target tile
  float* bt = smem + 32 * TSTRIDE;                 // 16 x 260 context tile
  float* d2 = smem + 32 * TSTRIDE + 16 * TSTRIDE;  // 32 x 20 score tile

  const int lane = threadIdx.x & 31;
  const int wid = blockIdx.x;            // 0..511 row tiles
  const int chunk = blockIdx.y;          // 0..CHUNKS-1
  const int b = wid / (NT / 32);
  const int row0 = (wid % (NT / 32)) * 32;
  const int ct0 = chunk * (NC / 16 / CHUNKS);
  const int ct1 = ct0 + (NC / 16 / CHUNKS);

  const float* xtb = xt + ((long)b * NT + row0) * 256;
  const float* xcb = xc + (long)b * NC * 256;
  const float* ncb = nc + (long)b * NC;

  stage_tile(at, xtb, 32, lane);

  float best[KNN];
  int bidx[KNN];
#pragma unroll
  for (int t = 0; t < KNN; ++t) { best[t] = 3.0e38f; bidx[t] = 0; }

  const int hw = lane >> 4, lo = lane & 15;
  const int off0 = lo * TSTRIDE + hw * 2;
  const int off1 = (lo + 16) * TSTRIDE + hw * 2;

  for (int ct = ct0; ct < ct1; ++ct) {
    if (ct + 1 < ct1) {  // hint next 16KB tile toward the caches
      __builtin_prefetch(xcb + (long)(ct + 1) * 16 * 256 + lane * 16, 0, 1);
      __builtin_prefetch(xcb + (long)(ct + 1) * 16 * 256 + 2048 + lane * 16, 0, 1);
    }
    stage_tile(bt, xcb + (long)ct * 16 * 256, 16, lane);

    v8f c0 = {}, c1 = {};
#pragma unroll 8
    for (int k = 0; k < 256; k += 4) {
      v2f bv = *(const v2f*)(bt + off0 + k);  // B[k][n=lo] = Xc[lo][k]
      v2f a0 = *(const v2f*)(at + off0 + k);
      v2f a1 = *(const v2f*)(at + off1 + k);
      c0 = wmma4(a0, bv, c0);
      c1 = wmma4(a1, bv, c1);
    }

    const float ncv = ncb[ct * 16 + lo];  // column norm (col = lo for all vgprs)
#pragma unroll
    for (int r = 0; r < 8; ++r) {
      const int m = r + hw * 8;
      d2[m * 20 + lo] = ncv - 2.0f * c0[r];
      d2[(m + 16) * 20 + lo] = ncv - 2.0f * c1[r];
    }

    // Each lane owns one of the 32 rows; scan this tile's 16 candidates.
    const float* myrow = d2 + lane * 20;
    const int base = ct * 16;
#pragma unroll
    for (int q = 0; q < 4; ++q) {
      float4 vv = *(const float4*)(myrow + q * 4);
      topk_insert(vv.x, base + q * 4 + 0, best, bidx);
      topk_insert(vv.y, base + q * 4 + 1, best, bidx);
      topk_insert(vv.z, base + q * 4 + 2, best, bidx);
      topk_insert(vv.w, base + q * 4 + 3, best, bidx);
    }
  }

  // Partial lists: [row][chunk][8], value-sorted ascending.
  const long pbase = (((long)b * NT + row0 + lane) * CHUNKS + chunk) * KNN;
#pragma unroll
  for (int t = 0; t < KNN; ++t) {
    pval[pbase + t] = best[t];
    pidx[pbase + t] = bidx[t];
  }
}

// Merge CHUNKS*8 sorted candidates per row into the final top-8.
// Chunk order == ascending index ranges, so stable strict-< insertion keeps
// lowest-index-first on ties.  One thread per target row.
__global__ __launch_bounds__(256) void knn_merge_kernel(
    const float* __restrict__ pval, const int* __restrict__ pidx,
    int* __restrict__ knn) {
  const int row = blockIdx.x * 256 + threadIdx.x;  // < B*NT
  float best[KNN];
  int bidx[KNN];
#pragma unroll
  for (int t = 0; t < KNN; ++t) { best[t] = 3.0e38f; bidx[t] = 0; }
  const float* v = pval + (long)row * CHUNKS * KNN;
  const int* ii = pidx + (long)row * CHUNKS * KNN;
  for (int t = 0; t < CHUNKS * KNN; ++t) topk_insert(v[t], ii[t], best, bidx);
  int* kout = knn + (long)row * KNN;
#pragma unroll
  for (int t = 0; t < KNN; ++t) kout[t] = bidx[t];
}

// ---------------------------------------------------------------------------
// Edge build: X_edge[i] = (Xn_t[i] + sum_j Xn_c[knn[i][j]]) / 9.
// ---------------------------------------------------------------------------
__global__ __launch_bounds__(256) void edge_gather_kernel(
    const float* __restrict__ xnt, const float* __restrict__ xnc,
    const int* __restrict__ knn, float* __restrict__ xedge) {
  const int i = blockIdx.x;  // 0..B*NT-1
  const int b = i >> 12;
  const int c = threadIdx.x;
  __shared__ int sidx[KNN];
  if (c < KNN) sidx[c] = knn[(long)i * KNN + c];
  __syncthreads();
  float acc = xnt[(long)i * 256 + c];
  const float* cb = xnc + (long)b * NC * 256;
#pragma unroll
  for (int j = 0; j < KNN; ++j) acc += cb[(long)sidx[j] * 256 + c];
  xedge[(long)i * 256 + c] = acc * (1.0f / 9.0f);
}

// Scatter edge messages back to context nodes (+degree counts).
__global__ __launch_bounds__(256) void scatter_kernel(
    const float* __restrict__ xet, const int* __restrict__ knn,
    float* __restrict__ contrib, float* __restrict__ counts) {
  const int i = blockIdx.x;
  const int b = i >> 12;
  const int c = threadIdx.x;
  __shared__ int sidx[KNN];
  if (c < KNN) sidx[c] = knn[(long)i * KNN + c];
  __syncthreads();
  const float v = xet[(long)i * 256 + c];
  float* cb = contrib + (long)b * NC * 256;
#pragma unroll
  for (int j = 0; j < KNN; ++j) atomicAdd(cb + (long)sidx[j] * 256 + c, v);
  if (c < KNN) atomicAdd(counts + (long)b * NC + sidx[c], 1.0f);
}

__global__ __launch_bounds__(256) void fill_zero_kernel(float* __restrict__ p,
                                                        long n) {
  long i = (long)blockIdx.x * blockDim.x + threadIdx.x;
  const long stride = (long)gridDim.x * blockDim.x;
  for (; i < n; i += stride) p[i] = 0.0f;
}

// ---------------------------------------------------------------------------
extern "C" void kernel_launch(void* const* d_in, const int* in_sizes, int n_in,
                              void* d_out, int out_size, void* d_ws,
                              size_t ws_size, hipStream_t stream) {
  (void)in_sizes; (void)n_in; (void)out_size; (void)ws_size;
  const float* Xt_img = (const float*)d_in[0];
  const float* Xc1_img = (const float*)d_in[1];
  const float* Xc2_img = (const float*)d_in[2];
  const float* W1 = (const float*)d_in[3];
  const float* b1 = (const float*)d_in[4];
  const float* W2 = (const float*)d_in[5];
  const float* b2 = (const float*)d_in[6];
  float* out = (float*)d_out;
  float* ws = (float*)d_ws;

  // Workspace layout (float offsets), ~140 MB total, 16B aligned throughout.
  const long SZ_TN = (long)BATCH * NT * CCH;   // 4,194,304
  const long SZ_CN = (long)BATCH * NC * CCH;   // 8,388,608
  float* xt   = ws;                    // target nodes  (reused as X_edge later)
  float* xc   = xt + SZ_TN;            // context nodes
  float* w1t  = xc + SZ_CN;            // W1^T [c_out][c_in]
  float* w2t  = w1t + CCH * CCH;       // W2^T
  float* nc   = w2t + CCH * CCH;       // context sq-norms
  float* xnt  = nc + (long)BATCH * NC; // node2edge(target)
  float* xnc  = xnt + SZ_TN;           // node2edge(context) (reused as X_et)
  int*   knnp = (int*)(xnc + SZ_CN);   // [B*NT][8]
  float* contrib = (float*)knnp + (long)BATCH * NT * KNN;
  float* counts  = contrib + SZ_CN;    // contiguous with contrib (one memfill)
  float* pval = counts + (long)BATCH * NC;           // [B*NT][CHUNKS][8]
  int*   pidx = (int*)(pval + (long)BATCH * NT * CHUNKS * KNN);
  float* xedge = xt;                   // alias: Xt dead after knn
  float* xet   = xnc;                  // alias: Xn_c dead after gather

  const dim3 tb(32, 8);
  const dim3 gX(NT / 32, CCH / 32, BATCH);

  // Layout: images/weights -> node-major rows.
  img2node_kernel<<<gX, tb, 0, stream>>>(Xt_img, xt, CCH, NT,
                                         (long)CCH * NT, (long)NT * CCH);
  img2node_kernel<<<gX, tb, 0, stream>>>(Xc1_img, xc, CCH, NT,
                                         (long)CCH * NT, (long)NC * CCH);
  img2node_kernel<<<gX, tb, 0, stream>>>(Xc2_img, xc + (long)NT * CCH, CCH, NT,
                                         (long)CCH * NT, (long)NC * CCH);
  img2node_kernel<<<dim3(8, 8, 1), tb, 0, stream>>>(W1, w1t, CCH, CCH,
                                                    CCH * CCH, CCH * CCH);
  img2node_kernel<<<dim3(8, 8, 1), tb, 0, stream>>>(W2, w2t, CCH, CCH,
                                                    CCH * CCH, CCH * CCH);

  norms_kernel<<<BATCH * NC / 8, 256, 0, stream>>>(xc, nc, BATCH * NC);

  // node2edge linear on all nodes (f32 WMMA GEMM).
  gemm_bias_kernel<<<dim3(BATCH * NT / 32, 16), 32, 0, stream>>>(xt, w1t, b1, xnt);
  gemm_bias_kernel<<<dim3(BATCH * NC / 32, 16), 32, 0, stream>>>(xc, w1t, b1, xnc);

  // kNN via WMMA distance sweep (8-way split for occupancy) + merge.
  knn_sweep_kernel<<<dim3(BATCH * NT / 32, CHUNKS), 32, 0, stream>>>(
      xt, xc, nc, pval, pidx);
  knn_merge_kernel<<<BATCH * NT / 256, 256, 0, stream>>>(pval, pidx, knnp);

  // Zero accumulators every launch (graph-replay safe).
  fill_zero_kernel<<<4096, 256, 0, stream>>>(contrib, SZ_CN + (long)BATCH * NC);

  edge_gather_kernel<<<BATCH * NT, 256, 0, stream>>>(xnt, xnc, knnp, xedge);
  gemm_bias_kernel<<<dim3(BATCH * NT / 32, 16), 32, 0, stream>>>(xedge, w2t, b2, xet);
  scatter_kernel<<<BATCH * NT, 256, 0, stream>>>(xet, knnp, contrib, counts);

  // Outputs: out_t | out_c1 | out_c2, each [B,C,H,W].
  node2img_kernel<<<gX, tb, 0, stream>>>(xet, out, nullptr,
                                         (long)NT * CCH, (long)CCH * NT, 0);
  node2img_kernel<<<gX, tb, 0, stream>>>(contrib, out + SZ_TN, counts,
                                         (long)NC * CCH, (long)CCH * NT, NC);
  node2img_kernel<<<gX, tb, 0, stream>>>(contrib + (long)NT * CCH,
                                         out + 2 * SZ_TN, counts + NT,
                                         (long)NC * CCH, (long)CCH * NT, NC);
}